// NegativeSampler_19937238188801
// MI455X (gfx1250) — compile-verified
//
#include <hip/hip_runtime.h>
#include <hip/hip_bf16.h>
#include <stdint.h>

#define VOCAB   50257
#define PAD     50260            // VOCAB rounded up to 4 floats (16B chunks)
#define LUTN    16384            // u-quantile bucket count (power of two -> exact q)
#define LUTPAD  16388            // LUTN+1 entries rounded up to 4 ints
#define TOTF    (PAD + LUTPAD)   // staged 32-bit words: cdf then lut, contiguous in ws
#define NCH     (TOTF / 4)       // 16662 16-byte async-DMA chunks
#define SCAN_T  1024
#define SAMP_T  1024
#define SAMP_B  512
#define EXPN    0.75f

// ---------------------------------------------------------------------------
// Kernel 1: total = sum(f); probs = (f/total)^0.75 with last token zeroed;
// inclusive prefix scan -> cdf[PAD] in workspace (pads replicate cdf[-1]).
// ---------------------------------------------------------------------------
__global__ __launch_bounds__(SCAN_T)
void build_cdf_kernel(const float* __restrict__ freq, float* __restrict__ cdf)
{
    __shared__ float buf[SCAN_T];
    const int tid = threadIdx.x;

    float s = 0.f;
    for (int i = tid; i < VOCAB; i += SCAN_T) s += freq[i];
    buf[tid] = s;
    __syncthreads();
    for (int off = SCAN_T >> 1; off > 0; off >>= 1) {
        if (tid < off) buf[tid] += buf[tid + off];
        __syncthreads();
    }
    const float inv = 1.0f / buf[0];
    __syncthreads();

    float carry = 0.f;
    for (int base = 0; base < PAD; base += SCAN_T) {
        const int i = base + tid;
        float v = 0.f;
        if (i < VOCAB - 1) v = powf(freq[i] * inv, EXPN);  // last token & pads = 0
        buf[tid] = v;
        __syncthreads();
        for (int off = 1; off < SCAN_T; off <<= 1) {
            const float t = (tid >= off) ? buf[tid - off] : 0.f;
            __syncthreads();
            buf[tid] += t;
            __syncthreads();
        }
        if (i < PAD) cdf[i] = carry + buf[tid];
        carry += buf[SCAN_T - 1];
        __syncthreads();
    }
}

// ---------------------------------------------------------------------------
// Kernel 2: lut[q] = searchsorted(cdf, (q/LUTN)*total, 'right'), q in [0,LUTPAD).
// Branchless 16-step search over the (L2-resident) global cdf; index probes
// clamped with min() so no OOB global reads. lut[LUTN..] saturates to VOCAB.
// ---------------------------------------------------------------------------
__global__ __launch_bounds__(256)
void build_lut_kernel(const float* __restrict__ cdf, int* __restrict__ lut)
{
    const int q = blockIdx.x * 256 + threadIdx.x;
    if (q >= LUTPAD) return;
    const float total = cdf[VOCAB - 1];
    const int   qq    = q < LUTN ? q : LUTN;
    const float L     = ((float)qq * (1.0f / (float)LUTN)) * total; // exact q/LUTN

    int p = 0;                       // p = #elements <= L
#pragma unroll
    for (int step = 1 << 15; step > 0; step >>= 1) {
        const int c   = p + step;
        const int idx = (c < VOCAB ? c : VOCAB) - 1;     // clamped probe, in-range
        const float v = cdf[idx];
        p = (c <= VOCAB && v <= L) ? c : p;
    }
    lut[q] = p;
}

// ---------------------------------------------------------------------------
// Threefry-2x32 (20 rounds), key = jax.random.key(1) -> (k0,k1) = (0,1).
// ---------------------------------------------------------------------------
__device__ __forceinline__ uint32_t rotl32(uint32_t x, uint32_t r) {
    return (x << r) | (x >> (32u - r));
}

__device__ __forceinline__ void threefry2x32(uint32_t c0, uint32_t c1,
                                             uint32_t& o0, uint32_t& o1)
{
    const uint32_t k0 = 0u, k1 = 1u;
    const uint32_t k2 = 0x1BD11BDAu ^ k0 ^ k1;
    uint32_t x0 = c0 + k0, x1 = c1 + k1;
#define TF_R(r) { x0 += x1; x1 = rotl32(x1, r); x1 ^= x0; }
    TF_R(13) TF_R(15) TF_R(26) TF_R(6)   x0 += k1; x1 += k2 + 1u;
    TF_R(17) TF_R(29) TF_R(16) TF_R(24)  x0 += k2; x1 += k0 + 2u;
    TF_R(13) TF_R(15) TF_R(26) TF_R(6)   x0 += k0; x1 += k1 + 3u;
    TF_R(17) TF_R(29) TF_R(16) TF_R(24)  x0 += k1; x1 += k2 + 4u;
    TF_R(13) TF_R(15) TF_R(26) TF_R(6)   x0 += k2; x1 += k0 + 5u;
#undef TF_R
    o0 = x0; o1 = x1;
}

__device__ __forceinline__ float bits_to_unit(uint32_t b) {
    return __uint_as_float((b >> 9) | 0x3f800000u) - 1.0f;   // [0,1)
}

// LUT-narrowed searchsorted(cdf, u, 'right'): answer in [lut[q], lut[q+1]],
// average window ~3 tokens -> ~2 LDS probes.
__device__ __forceinline__ int search_lut(const float* __restrict__ cdfL,
                                          const int* __restrict__ lutL,
                                          float unit, float total)
{
    const float u = unit * total;
    const int   q = (int)(unit * (float)LUTN);   // exact: power-of-2 scale
    int lo = lutL[q];
    int hi = lutL[q + 1];
    while (lo < hi) {
        const int mid = (lo + hi) >> 1;
        if (cdfL[mid] > u) hi = mid; else lo = mid + 1;
    }
    return lo < VOCAB ? lo : VOCAB - 1;
}

// ---------------------------------------------------------------------------
// Kernel 3: async-DMA the 201KB CDF + 64KB LUT (contiguous in ws) into the
// 320KB LDS, then inverse-CDF sample. One block per WGP, 32 wave32s to hide
// the non-temporal store stream (134MB -> ~5.8us HBM floor).
// ---------------------------------------------------------------------------
__global__ __launch_bounds__(SAMP_T)
void sample_kernel(const float* __restrict__ ws, int* __restrict__ out, long long n)
{
    extern __shared__ float lds[];               // [PAD] cdf | [LUTPAD] lut
    const int tid = threadIdx.x;

    {   // ---- single async staging loop: ws is laid out exactly like LDS ----
        const uint32_t lds_base = (uint32_t)(uintptr_t)(&lds[0]);
        for (int c = tid; c < NCH; c += SAMP_T) {
            const float*   g  = ws + 4 * c;
            const uint32_t la = lds_base + 16u * (uint32_t)c;
            asm volatile("global_load_async_to_lds_b128 %0, %1, off"
                         :: "v"(la), "v"(g) : "memory");
        }
        asm volatile("s_wait_asynccnt 0x0" ::: "memory");    // ASYNCcnt == 0
    }
    __syncthreads();

    const float* cdfL = lds;
    const int*   lutL = (const int*)(lds + PAD);
    const float  total = cdfL[VOCAB - 1];

    const long long half   = n >> 1;
    const long long stride = (long long)gridDim.x * SAMP_T;

    for (long long j = (long long)blockIdx.x * SAMP_T + tid; j < half; j += stride) {
        uint32_t b0, b1;
        threefry2x32((uint32_t)j, (uint32_t)(j + half), b0, b1); // JAX counter split
        const int s0 = search_lut(cdfL, lutL, bits_to_unit(b0), total);
        const int s1 = search_lut(cdfL, lutL, bits_to_unit(b1), total);
        __builtin_nontemporal_store(s0, &out[j]);
        __builtin_nontemporal_store(s1, &out[j + half]);
    }
}

// ---------------------------------------------------------------------------
extern "C" void kernel_launch(void* const* d_in, const int* in_sizes, int n_in,
                              void* d_out, int out_size, void* d_ws, size_t ws_size,
                              hipStream_t stream)
{
    (void)in_sizes; (void)n_in; (void)ws_size;
    const float* freq = (const float*)d_in[0];
    float*       cdf  = (float*)d_ws;                  // PAD floats
    int*         lut  = (int*)((float*)d_ws + PAD);    // LUTPAD ints, contiguous
    int*         out  = (int*)d_out;
    const long long n = (long long)out_size;           // 33,554,432

    build_cdf_kernel<<<1, SCAN_T, 0, stream>>>(freq, cdf);
    build_lut_kernel<<<(LUTPAD + 255) / 256, 256, 0, stream>>>(cdf, lut);
    sample_kernel<<<SAMP_B, SAMP_T, TOTF * sizeof(float), stream>>>(
        (const float*)d_ws, out, n);
}